// StackedGCNMeetupV2_72464688218153
// MI455X (gfx1250) — compile-verified
//
#include <hip/hip_runtime.h>

// CDNA5 / gfx1250. wave32. f32 end-to-end (reference is f32; FLOPs are tiny,
// the workload is L2-atomic bound, so no reason to drop precision).
// Dense x@W0 (Nx8 @ 8x16) runs on V_WMMA_F32_16X16X4_F32, two K=4 steps.

typedef __attribute__((ext_vector_type(2))) float v2f;
typedef __attribute__((ext_vector_type(8))) float v8f;

static constexpr int Ucap = 200000;
static constexpr int Tcap = 20000;
static constexpr int Ccap = 5000;
static constexpr int Gcap = 10000;

__device__ __forceinline__ int clampi(int v, int lo, int hi) {
  return v < lo ? lo : (v > hi ? hi : v);
}

// deg=1 (self loop), agg1 = b0 broadcast, out = b2 broadcast
__global__ void k_init(float* __restrict__ deg, float* __restrict__ agg1,
                       float* __restrict__ out, const float* __restrict__ b0,
                       const float* __restrict__ b2, int n) {
  int t = blockIdx.x * blockDim.x + threadIdx.x;
  if (t < n * 16) agg1[t] = b0[t & 15];
  if (t < n) { deg[t] = 1.0f; out[t] = b2[0]; }
}

__global__ void k_degree(const int* __restrict__ col, float* __restrict__ deg, int e) {
  int t = blockIdx.x * blockDim.x + threadIdx.x;
  if (t < e) atomicAdd(&deg[col[t]], 1.0f);
}

__global__ void k_rsqrt(float* __restrict__ deg, int n) {
  int t = blockIdx.x * blockDim.x + threadIdx.x;
  if (t < n) deg[t] = rsqrtf(deg[t]);   // deg >= 1 always (self loop)
}

__global__ void k_assemble(const int* __restrict__ feat, const int* __restrict__ lmask,
                           const float* __restrict__ user_emb, const float* __restrict__ known_emb,
                           const float* __restrict__ mask_emb, const float* __restrict__ cat_emb,
                           const float* __restrict__ topic_emb, const float* __restrict__ group_emb,
                           const float* __restrict__ user_W, const float* __restrict__ user_b,
                           const float* __restrict__ mask_W, const float* __restrict__ mask_b,
                           const float* __restrict__ cat_W, const float* __restrict__ cat_b,
                           const float* __restrict__ topic_W, const float* __restrict__ topic_b,
                           const float* __restrict__ group_W, const float* __restrict__ group_b,
                           float* __restrict__ xout, int n) {
  int i = blockIdx.x * blockDim.x + threadIdx.x;
  if (i >= n) return;
  int idx = feat[i * 3 + 0];
  int known = feat[i * 3 + 1];
  int tc = feat[i * 3 + 2];

  float ebuf[8];
  const float* Wm = nullptr;
  const float* bm = nullptr;
  int D = 0;
  if (tc == 0) {
    int id = clampi(idx, 0, Ucap - 1);
    int kn = clampi(known, 0, 1);
#pragma unroll
    for (int d = 0; d < 8; d++) ebuf[d] = user_emb[id * 8 + d] + known_emb[kn * 8 + d];
    Wm = user_W; bm = user_b; D = 8;
  } else if (tc == 1) {
    int id = clampi(idx, 0, Tcap - 1);
#pragma unroll
    for (int d = 0; d < 8; d++) ebuf[d] = topic_emb[id * 8 + d];
    Wm = topic_W; bm = topic_b; D = 8;
  } else if (tc == 2) {
    int id = clampi(idx, 0, Ccap - 1);
    ebuf[0] = cat_emb[id * 2 + 0];
    ebuf[1] = cat_emb[id * 2 + 1];
    Wm = cat_W; bm = cat_b; D = 2;
  } else if (tc == 4) {
    int id = clampi(idx, 0, Gcap - 1);
#pragma unroll
    for (int d = 0; d < 8; d++) ebuf[d] = group_emb[id * 8 + d];
    Wm = group_W; bm = group_b; D = 8;
  }

  float f[8];
  if (D) {
#pragma unroll
    for (int c = 0; c < 8; c++) {
      float s = bm[c];
      for (int d = 0; d < D; d++) s += fmaxf(ebuf[d], 0.0f) * Wm[d * 8 + c];
      f[c] = s;
    }
  } else {
#pragma unroll
    for (int c = 0; c < 8; c++) f[c] = 0.0f;
  }

  int lm = lmask[i] ? 1 : 0;
  float m[8];
#pragma unroll
  for (int d = 0; d < 8; d++) m[d] = fmaxf(mask_emb[lm * 8 + d], 0.0f);
#pragma unroll
  for (int c = 0; c < 8; c++) {
    float s = mask_b[c];
#pragma unroll
    for (int d = 0; d < 8; d++) s += m[d] * mask_W[d * 8 + c];
    float mf = 1.0f / (1.0f + expf(-s));
    xout[i * 8 + c] = f[c] * mf;
  }
}

// h = x @ W0 : per-wave 16-row tile; two V_WMMA_F32_16X16X4_F32 (K=0..3, 4..7).
// A/B VGPR layout (32-bit): element(lane, vgpr) -> M|N = lane%16, K = (lane/16)*2 + vgpr.
// D layout: vgpr j, lane L -> row M = (L/16)*8 + j, col N = L%16.
__global__ void k_xw0_wmma(const float* __restrict__ x, const float* __restrict__ W0,
                           float* __restrict__ h, int ntiles) {
  int wave = (blockIdx.x * blockDim.x + threadIdx.x) >> 5;
  if (wave >= ntiles) return;            // uniform per wave -> EXEC all-ones at WMMA
  int lane = threadIdx.x & 31;
  int half = lane >> 4;
  int m = lane & 15;
  int k0 = half * 2;

  const float* xr = x + (size_t)(wave * 16 + m) * 8;
  v2f a0, a1, b0, b1;
  a0.x = xr[k0];     a0.y = xr[k0 + 1];
  a1.x = xr[4 + k0]; a1.y = xr[5 + k0];
  b0.x = W0[(k0) * 16 + m];     b0.y = W0[(k0 + 1) * 16 + m];
  b1.x = W0[(4 + k0) * 16 + m]; b1.y = W0[(5 + k0) * 16 + m];

  v8f c = {};
  c = __builtin_amdgcn_wmma_f32_16x16x4_f32(false, a0, false, b0, (short)0, c, false, false);
  c = __builtin_amdgcn_wmma_f32_16x16x4_f32(false, a1, false, b1, (short)0, c, false, false);

  float* hr = h + (size_t)(wave * 16 + half * 8) * 16 + m;
#pragma unroll
  for (int j = 0; j < 8; j++) hr[j * 16] = c[j];
}

// Layer-1 aggregation: items [0,e) are edges, [e, e+n) are self loops.
// 4 threads per item, each moves a float4 of the 16-wide message.
__global__ void k_agg16(const int* __restrict__ row, const int* __restrict__ col,
                        const float* __restrict__ dis, const float* __restrict__ h,
                        float* __restrict__ agg, int e, int n) {
  int t = blockIdx.x * blockDim.x + threadIdx.x;
  int item = t >> 2;
  int q = t & 3;
  if (item >= e + n) return;
  int r, cl;
  if (item < e) { r = row[item]; cl = col[item]; }
  else          { r = item - e; cl = r; }
  float nrm = dis[r] * dis[cl];
  const float4 v = *(const float4*)(h + (size_t)r * 16 + q * 4);
  float* o = agg + (size_t)cl * 16 + q * 4;
  atomicAdd(o + 0, nrm * v.x);
  atomicAdd(o + 1, nrm * v.y);
  atomicAdd(o + 2, nrm * v.z);
  atomicAdd(o + 3, nrm * v.w);
}

// relu(agg1) @ W2 (16x1) -> h2
__global__ void k_l2(const float* __restrict__ agg, const float* __restrict__ W2,
                     float* __restrict__ h2, int n) {
  int i = blockIdx.x * blockDim.x + threadIdx.x;
  if (i >= n) return;
  const float* a = agg + (size_t)i * 16;
  float s = 0.0f;
#pragma unroll
  for (int j = 0; j < 16; j++) s += fmaxf(a[j], 0.0f) * W2[j];
  h2[i] = s;
}

// Layer-2 aggregation: scalar messages into d_out (pre-seeded with b2).
__global__ void k_agg1(const int* __restrict__ row, const int* __restrict__ col,
                       const float* __restrict__ dis, const float* __restrict__ h2,
                       float* __restrict__ out, int e, int n) {
  int item = blockIdx.x * blockDim.x + threadIdx.x;
  if (item >= e + n) return;
  int r, cl;
  if (item < e) { r = row[item]; cl = col[item]; }
  else          { r = item - e; cl = r; }
  atomicAdd(&out[cl], dis[r] * dis[cl] * h2[r]);
}

extern "C" void kernel_launch(void* const* d_in, const int* in_sizes, int n_in,
                              void* d_out, int out_size, void* d_ws, size_t ws_size,
                              hipStream_t stream) {
  const int* edges = (const int*)d_in[0];
  const int e = in_sizes[0] / 2;
  const int* feat  = (const int*)d_in[1];
  const int* lmask = (const int*)d_in[2];
  const int n = in_sizes[2];

  const float* user_emb  = (const float*)d_in[3];
  const float* known_emb = (const float*)d_in[4];
  const float* mask_emb  = (const float*)d_in[5];
  const float* cat_emb   = (const float*)d_in[6];
  const float* topic_emb = (const float*)d_in[7];
  const float* group_emb = (const float*)d_in[8];
  const float* user_W  = (const float*)d_in[9];
  const float* user_b  = (const float*)d_in[10];
  const float* mask_W  = (const float*)d_in[11];
  const float* mask_b  = (const float*)d_in[12];
  const float* cat_W   = (const float*)d_in[13];
  const float* cat_b   = (const float*)d_in[14];
  const float* topic_W = (const float*)d_in[15];
  const float* topic_b = (const float*)d_in[16];
  const float* group_W = (const float*)d_in[17];
  const float* group_b = (const float*)d_in[18];
  const float* W0 = (const float*)d_in[19];
  const float* b0 = (const float*)d_in[20];
  const float* W2 = (const float*)d_in[21];
  const float* b2 = (const float*)d_in[22];

  const int* row = edges;
  const int* col = edges + e;

  // workspace carve-up (16-float aligned slots); npad rows for WMMA tiles
  const int npad = (n + 15) & ~15;
  auto align16 = [](size_t v) { return (v + 15) & ~(size_t)15; };
  float* ws = (float*)d_ws;
  size_t off = 0;
  float* dis  = ws + off; off = align16(off + (size_t)n);
  float* xb   = ws + off; off = align16(off + (size_t)npad * 8);
  float* hb   = ws + off; off = align16(off + (size_t)npad * 16);
  float* agg1 = ws + off; off = align16(off + (size_t)n * 16);
  float* h2   = ws + off; off = align16(off + (size_t)n);
  float* out = (float*)d_out;

  const int B = 256;
  k_init<<<(n * 16 + B - 1) / B, B, 0, stream>>>(dis, agg1, out, b0, b2, n);
  k_degree<<<(e + B - 1) / B, B, 0, stream>>>(col, dis, e);
  k_rsqrt<<<(n + B - 1) / B, B, 0, stream>>>(dis, n);
  k_assemble<<<(n + B - 1) / B, B, 0, stream>>>(
      feat, lmask, user_emb, known_emb, mask_emb, cat_emb, topic_emb, group_emb,
      user_W, user_b, mask_W, mask_b, cat_W, cat_b, topic_W, topic_b,
      group_W, group_b, xb, n);

  const int ntiles = npad / 16;   // one wave per 16-row tile
  k_xw0_wmma<<<(ntiles * 32 + B - 1) / B, B, 0, stream>>>(xb, W0, hb, ntiles);

  const long long items4 = (long long)(e + n) * 4;
  k_agg16<<<(int)((items4 + B - 1) / B), B, 0, stream>>>(row, col, dis, hb, agg1, e, n);
  k_l2<<<(n + B - 1) / B, B, 0, stream>>>(agg1, W2, h2, n);
  k_agg1<<<(e + n + B - 1) / B, B, 0, stream>>>(row, col, dis, h2, out, e, n);
}